// sub_abs_64939905516355
// MI455X (gfx1250) — compile-verified
//
#include <hip/hip_runtime.h>
#include <math.h>

// Problem constants from the reference (Q taken from in_sizes at launch).
#define C_N    64
#define SHOT_N 5
#define D_N    512

#define TQ      8              // queries per wave
#define WAVES   4              // waves per block (wave32)
#define BLOCK_T (WAVES * 32)   // 128 threads
#define QB      (WAVES * TQ)   // 32 queries per block
#define LDS_FLOATS (C_N * D_N) // protoT[d][c] : 32768 floats = 128 KB

// ---------------------------------------------------------------------------
// Kernel 1: proto^T[d][c] = mean over shots of y[c][s][d], written transposed
// so the main kernel's LDS fill is one contiguous 128 KB async copy and LDS
// reads are bank-conflict-free (bank = lane).
// ---------------------------------------------------------------------------
__global__ void proto_mean_kernel(const float* __restrict__ y,
                                  float* __restrict__ protoT) {
  int t = blockIdx.x * blockDim.x + threadIdx.x;
  if (t >= C_N * D_N) return;
  int c = t & (C_N - 1);   // fast index -> coalesced writes to protoT
  int d = t >> 6;
  const float* yc = y + (size_t)c * (SHOT_N * D_N) + d;
  float s = yc[0] + yc[D_N] + yc[2 * D_N] + yc[3 * D_N] + yc[4 * D_N];
  protoT[t] = s * (1.0f / SHOT_N);   // t == d*64 + c
}

// ---------------------------------------------------------------------------
// Kernel 2: L1 distances + softmax. lane = class (lane, lane+32);
// each wave accumulates dist for TQ queries across all 64 classes.
// x rows are wave-uniform -> readfirstlane makes that provable so the
// backend issues SMEM (s_load) for x, keeping the vector pipe for DS.
// ---------------------------------------------------------------------------
template <bool USE_WS>
__global__ __launch_bounds__(BLOCK_T) void protonet_l1_softmax(
    const float* __restrict__ x,
    const float* __restrict__ y,
    const float* __restrict__ protoT,
    float* __restrict__ out,
    int Q) {
  extern __shared__ float lds[];  // protoT tile: [D_N][C_N]
  const int tid  = threadIdx.x;
  const int lane = tid & 31;
  const int wave = tid >> 5;

  if (USE_WS) {
    // CDNA5 async tensor-data path: global -> LDS, b128 per lane,
    // tracked on ASYNCcnt (does not touch the VGPR data path).
    unsigned lds_base = (unsigned)(size_t)(&lds[0]);  // low 32 bits = LDS offset
    const char* gbase = (const char*)protoT;
    #pragma unroll 1
    for (int i = tid; i < LDS_FLOATS / 4; i += BLOCK_T) {  // 8192 16B chunks
      unsigned    laddr = lds_base + (unsigned)(i * 16);
      const char* gaddr = gbase + (size_t)i * 16;
      asm volatile("global_load_async_to_lds_b128 %0, %1, off"
                   :: "v"(laddr), "v"(gaddr)
                   : "memory");
    }
#if __has_builtin(__builtin_amdgcn_s_wait_asynccnt)
    __builtin_amdgcn_s_wait_asynccnt(0);
#else
    asm volatile("s_wait_asynccnt 0x0" ::: "memory");
#endif
  } else {
    // Fallback: compute proto^T directly into LDS from y (L2-resident, tiny).
    for (int t = tid; t < C_N * D_N; t += BLOCK_T) {
      int c = t & (C_N - 1);
      int d = t >> 6;
      const float* yc = y + (size_t)c * (SHOT_N * D_N) + d;
      float s = yc[0] + yc[D_N] + yc[2 * D_N] + yc[3 * D_N] + yc[4 * D_N];
      lds[t] = s * (1.0f / SHOT_N);
    }
  }
  __syncthreads();

  // Wave-uniform query base; readfirstlane makes uniformity provable so x
  // row loads select onto the scalar (SMEM) path.
  const int q0 = __builtin_amdgcn_readfirstlane(blockIdx.x * QB + wave * TQ);

  float accA[TQ], accB[TQ];
  #pragma unroll
  for (int q = 0; q < TQ; ++q) { accA[q] = 0.0f; accB[q] = 0.0f; }

  // Per-query row pointers (uniform; clamped for ragged tails).
  const float* xrow[TQ];
  #pragma unroll
  for (int q = 0; q < TQ; ++q) {
    int qi = q0 + q;
    if (qi > Q - 1) qi = Q - 1;
    xrow[q] = x + (size_t)qi * D_N;
  }

  // Main loop: per 4-d chunk => 8 conflict-free ds_load_b32 + TQ uniform
  // (scalar) 16B x loads vs 128 VALU ops. abs folds into VOP3 |src|
  // modifiers; unroll x4 lets SMEM loads merge and VOPD pairs form.
  #pragma unroll 4
  for (int d = 0; d < D_N; d += 4) {
    float pA0 = lds[(d + 0) * C_N + lane];
    float pA1 = lds[(d + 1) * C_N + lane];
    float pA2 = lds[(d + 2) * C_N + lane];
    float pA3 = lds[(d + 3) * C_N + lane];
    float pB0 = lds[(d + 0) * C_N + 32 + lane];
    float pB1 = lds[(d + 1) * C_N + 32 + lane];
    float pB2 = lds[(d + 2) * C_N + 32 + lane];
    float pB3 = lds[(d + 3) * C_N + 32 + lane];
    #pragma unroll
    for (int q = 0; q < TQ; ++q) {
      const float4 xv = *(const float4*)(xrow[q] + d);  // wave-uniform -> s_load
      accA[q] += (fabsf(xv.x - pA0) + fabsf(xv.y - pA1)) +
                 (fabsf(xv.z - pA2) + fabsf(xv.w - pA3));
      accB[q] += (fabsf(xv.x - pB0) + fabsf(xv.y - pB1)) +
                 (fabsf(xv.z - pB2) + fabsf(xv.w - pB3));
    }
  }

  // Softmax over classes: cross-lane (wave32) butterflies.
  #pragma unroll 1
  for (int q = 0; q < TQ; ++q) {
    float dA = accA[q], dB = accB[q];
    float m = fminf(dA, dB);
    #pragma unroll
    for (int k = 16; k; k >>= 1) m = fminf(m, __shfl_xor(m, k, 32));
    // softmax(-dist): exp(m - d) with m = min dist
    float eA = __expf(m - dA);
    float eB = __expf(m - dB);
    float s = eA + eB;
    #pragma unroll
    for (int k = 16; k; k >>= 1) s += __shfl_xor(s, k, 32);
    float inv = 1.0f / s;
    int qi = q0 + q;
    if (qi < Q) {
      float* o = out + (size_t)qi * C_N;
      o[lane]      = eA * inv;
      o[lane + 32] = eB * inv;
    }
  }
}

// ---------------------------------------------------------------------------
extern "C" void kernel_launch(void* const* d_in, const int* in_sizes, int n_in,
                              void* d_out, int out_size, void* d_ws, size_t ws_size,
                              hipStream_t stream) {
  const float* x = (const float*)d_in[0];
  const float* y = (const float*)d_in[1];
  float* out = (float*)d_out;

  const int Q = in_sizes[0] / D_N;  // 8192 for the reference shapes
  const size_t protoBytes = (size_t)C_N * D_N * sizeof(float);
  const size_t shmem = (size_t)LDS_FLOATS * sizeof(float);  // 131072 B
  dim3 grid((Q + QB - 1) / QB);
  dim3 block(BLOCK_T);

  if (ws_size >= protoBytes) {
    float* protoT = (float*)d_ws;
    proto_mean_kernel<<<(C_N * D_N + 255) / 256, 256, 0, stream>>>(y, protoT);
    protonet_l1_softmax<true><<<grid, block, shmem, stream>>>(x, y, protoT, out, Q);
  } else {
    protonet_l1_softmax<false><<<grid, block, shmem, stream>>>(x, y, nullptr, out, Q);
  }
}